// LyricsLSTMAttention_63711544868995
// MI455X (gfx1250) — compile-verified
//
#include <hip/hip_runtime.h>

#define B_ 32
#define T_ 64
#define V_ 32000
#define E_ 300
#define H_ 512
#define NF_ 128
#define D_ 32
#define LSTM_IN_ 332
#define G4H_ 2048   // 4*H

typedef float v2f __attribute__((ext_vector_type(2)));
typedef float v8f __attribute__((ext_vector_type(8)));

__device__ __forceinline__ v8f wmma_k4(v2f a, v2f b, v8f c) {
    // D = A(16x4 f32) * B(4x16 f32) + C(16x16 f32)
    return __builtin_amdgcn_wmma_f32_16x16x4_f32(
        false, a, false, b, (short)0, c, false, false);
}

// C[16x16] tile at (m0,n0) of  A[M,K](row-major, stride lda)  @  W[N,K]^T (row-major, stride ldw)
// A-frag per lane: row = m0+(lane&15), cols kbase,kbase+1 with kbase=(lane>>4)*2 (+k)
// B-frag per lane: n  = n0+(lane&15), cols kbase,kbase+1 from W row n
__device__ __forceinline__ v8f tile_gemm(const float* __restrict__ A, int lda,
                                         const float* __restrict__ W, int ldw,
                                         int K, int m0, int n0, v8f acc) {
    const int lane = threadIdx.x & 31;
    const int kh   = (lane >> 4) * 2;
    const float* arow = A + (size_t)(m0 + (lane & 15)) * lda + kh;
    const float* wrow = W + (size_t)(n0 + (lane & 15)) * ldw + kh;
    for (int k = 0; k < K; k += 4) {
        v2f a = *(const v2f*)(arow + k);
        v2f b = *(const v2f*)(wrow + k);
        acc = wmma_k4(a, b, acc);
    }
    return acc;
}

__device__ __forceinline__ void store_tile(float* __restrict__ C, int ldc,
                                           int m0, int n0, v8f acc) {
    const int lane = threadIdx.x & 31;
    const int n  = n0 + (lane & 15);
    const int mb = m0 + (lane >> 4) * 8;
#pragma unroll
    for (int r = 0; r < 8; ++r) C[(size_t)(mb + r) * ldc + n] = acc[r];
}

__device__ __forceinline__ float sigmoidf_(float x) { return 1.f / (1.f + __expf(-x)); }

// ---------------- kernels ----------------

__global__ void k_zero(float* p, int n) {
    int i = blockIdx.x * blockDim.x + threadIdx.x;
    if (i < n) p[i] = 0.f;
}

__global__ void k_embed(const int* __restrict__ seq, const float* __restrict__ emb,
                        float* __restrict__ out) {
    int i = blockIdx.x * blockDim.x + threadIdx.x;   // B*T*E
    if (i >= B_ * T_ * E_) return;
    int e = i % E_, bt = i / E_;
    out[i] = emb[(size_t)seq[bt] * E_ + e];
}

// midi_proj[b,f,e] = relu( midi[b,f,:] . mp_w[e,:] + mp_b[e] )   M=B*NF, N=D, K=D
__global__ void k_midi_proj(const float* __restrict__ midi, const float* __restrict__ mp_w,
                            const float* __restrict__ mp_b, float* __restrict__ out) {
    int wave = (blockIdx.x * blockDim.x + threadIdx.x) >> 5;
    int nt = wave & 1;               // N/16 = 2
    int mt = wave >> 1;              // 256 tiles
    int m0 = mt * 16, n0 = nt * 16;
    v8f acc = {};
    acc = tile_gemm(midi, D_, mp_w, D_, D_, m0, n0, acc);
    const int lane = threadIdx.x & 31;
    const int n  = n0 + (lane & 15);
    const int mb = m0 + (lane >> 4) * 8;
    float bias = mp_b[n];
#pragma unroll
    for (int r = 0; r < 8; ++r) {
        float v = acc[r] + bias;
        out[(size_t)(mb + r) * D_ + n] = v > 0.f ? v : 0.f;
    }
}

// midi_key = midi_proj @ wm^T   M=B*NF, N=H, K=D
__global__ void k_midi_key(const float* __restrict__ mproj, const float* __restrict__ wm,
                           float* __restrict__ out) {
    int wave = (blockIdx.x * blockDim.x + threadIdx.x) >> 5;
    int nt = wave & 31;              // N/16 = 32
    int mt = wave >> 5;
    v8f acc = {};
    acc = tile_gemm(mproj, D_, wm, D_, D_, mt * 16, nt * 16, acc);
    store_tile(out, H_, mt * 16, nt * 16, acc);
}

// q = h_top @ wh^T   M=B, N=H, K=H
__global__ void k_q(const float* __restrict__ h1, const float* __restrict__ wh,
                    float* __restrict__ q) {
    int wave = (blockIdx.x * blockDim.x + threadIdx.x) >> 5;
    int nt = wave & 31;
    int mt = wave >> 5;              // 0..1
    v8f acc = {};
    acc = tile_gemm(h1, H_, wh, H_, H_, mt * 16, nt * 16, acc);
    store_tile(q, H_, mt * 16, nt * 16, acc);
}

// scores[b,f] = sum_h v[h] * tanh(q[b,h] + midi_key[b,f,h]) — one wave per (b,f)
__global__ void k_scores(const float* __restrict__ q, const float* __restrict__ mkey,
                         const float* __restrict__ vvec, float* __restrict__ scores) {
    int wave = (blockIdx.x * blockDim.x + threadIdx.x) >> 5;
    int lane = threadIdx.x & 31;
    int b = wave / NF_, f = wave % NF_;
    const float* qk = q + (size_t)b * H_;
    const float* mk = mkey + (size_t)(b * NF_ + f) * H_;
    float s = 0.f;
    for (int h = lane; h < H_; h += 32) s += vvec[h] * tanhf(qk[h] + mk[h]);
#pragma unroll
    for (int off = 16; off > 0; off >>= 1) s += __shfl_xor(s, off, 32);
    if (lane == 0) scores[b * NF_ + f] = s;
}

// softmax over NF + ctx = attn @ midi_proj ; one 128-thread block per batch row
__global__ void k_softmax_ctx(const float* __restrict__ scores, const float* __restrict__ mproj,
                              float* __restrict__ ctx, float* __restrict__ attn_out, int t) {
    __shared__ float sm[NF_];
    __shared__ float redmax[4], redsum[4];
    int b = blockIdx.x, f = threadIdx.x;
    float s = scores[b * NF_ + f];
    float m = s;
#pragma unroll
    for (int off = 16; off > 0; off >>= 1) m = fmaxf(m, __shfl_xor(m, off, 32));
    if ((f & 31) == 0) redmax[f >> 5] = m;
    __syncthreads();
    m = fmaxf(fmaxf(redmax[0], redmax[1]), fmaxf(redmax[2], redmax[3]));
    float e = __expf(s - m);
    float ss = e;
#pragma unroll
    for (int off = 16; off > 0; off >>= 1) ss += __shfl_xor(ss, off, 32);
    if ((f & 31) == 0) redsum[f >> 5] = ss;
    __syncthreads();
    float tot = redsum[0] + redsum[1] + redsum[2] + redsum[3];
    float a = e / tot;
    sm[f] = a;
    attn_out[(size_t)(b * T_ + t) * NF_ + f] = a;
    __syncthreads();
    if (f < D_) {
        float acc = 0.f;
        for (int ff = 0; ff < NF_; ++ff)
            acc += sm[ff] * mproj[(size_t)(b * NF_ + ff) * D_ + f];
        ctx[b * D_ + f] = acc;
    }
}

// gates0 = [emb_t | ctx] @ w_ih0^T + h0 @ w_hh0^T + b_ih0 + b_hh0   M=B, N=4H
__global__ void k_gates0(const float* __restrict__ embws, int t, const float* __restrict__ ctx,
                         const float* __restrict__ h0,
                         const float* __restrict__ wih, const float* __restrict__ whh,
                         const float* __restrict__ bih, const float* __restrict__ bhh,
                         float* __restrict__ gates) {
    int wave = (blockIdx.x * blockDim.x + threadIdx.x) >> 5;
    int nt = wave & 127;             // 4H/16 = 128
    int mt = wave >> 7;              // 0..1
    int m0 = mt * 16, n0 = nt * 16;
    const int lane = threadIdx.x & 31;
    const int n = n0 + (lane & 15);
    float bias = bih[n] + bhh[n];
    v8f acc;
#pragma unroll
    for (int r = 0; r < 8; ++r) acc[r] = bias;
    // segment 1: embedding slice x_t (rows stride T*E, offset t*E), cols 0..E-1 of wih
    acc = tile_gemm(embws + (size_t)t * E_, T_ * E_, wih, LSTM_IN_, E_, m0, n0, acc);
    // segment 2: ctx, cols E..E+D-1 of wih
    acc = tile_gemm(ctx, D_, wih + E_, LSTM_IN_, D_, m0, n0, acc);
    // segment 3: recurrent
    acc = tile_gemm(h0, H_, whh, H_, H_, m0, n0, acc);
    store_tile(gates, G4H_, m0, n0, acc);
}

// gates1 = h0new @ w_ih1^T + h1 @ w_hh1^T + biases
__global__ void k_gates1(const float* __restrict__ x, const float* __restrict__ h1,
                         const float* __restrict__ wih, const float* __restrict__ whh,
                         const float* __restrict__ bih, const float* __restrict__ bhh,
                         float* __restrict__ gates) {
    int wave = (blockIdx.x * blockDim.x + threadIdx.x) >> 5;
    int nt = wave & 127;
    int mt = wave >> 7;
    int m0 = mt * 16, n0 = nt * 16;
    const int lane = threadIdx.x & 31;
    const int n = n0 + (lane & 15);
    float bias = bih[n] + bhh[n];
    v8f acc;
#pragma unroll
    for (int r = 0; r < 8; ++r) acc[r] = bias;
    acc = tile_gemm(x,  H_, wih, H_, H_, m0, n0, acc);
    acc = tile_gemm(h1, H_, whh, H_, H_, m0, n0, acc);
    store_tile(gates, G4H_, m0, n0, acc);
}

// LSTM elementwise cell update (PyTorch gate order i,f,g,o); optionally emit h to lstm_out
__global__ void k_cell(const float* __restrict__ gates, float* __restrict__ h,
                       float* __restrict__ c, float* __restrict__ lstm_out, int t) {
    int j = blockIdx.x * blockDim.x + threadIdx.x;
    if (j >= B_ * H_) return;
    int b = j / H_, n = j % H_;
    const float* g = gates + (size_t)b * G4H_;
    float ig = sigmoidf_(g[n]);
    float fg = sigmoidf_(g[H_ + n]);
    float gg = tanhf(g[2 * H_ + n]);
    float og = sigmoidf_(g[3 * H_ + n]);
    float cn = fg * c[j] + ig * gg;
    float hn = og * tanhf(cn);
    c[j] = cn;
    h[j] = hn;
    if (lstm_out) lstm_out[(size_t)(b * T_ + t) * H_ + n] = hn;
}

__global__ void k_final_state(const float* __restrict__ h0, const float* __restrict__ c0,
                              const float* __restrict__ h1, const float* __restrict__ c1,
                              float* __restrict__ outF) {
    int j = blockIdx.x * blockDim.x + threadIdx.x;
    if (j >= B_ * H_) return;
    float* hF = outF;
    float* cF = outF + 2 * B_ * H_;
    hF[j] = h0[j];
    hF[B_ * H_ + j] = h1[j];
    cF[j] = c0[j];
    cF[B_ * H_ + j] = c1[j];
}

// FC head: [2048,512] @ fc_w[V,512]^T + fc_b ; 128x128 block tile, K-chunk 16, 8 waves
#define FC_BM 128
#define FC_BN 128
#define FC_BK 16
#define FC_PAD 20   // conflict-free + 16B aligned LDS rows
__global__ __launch_bounds__(256) void k_fc(const float* __restrict__ A,
                                            const float* __restrict__ W,
                                            const float* __restrict__ bias,
                                            float* __restrict__ Cout) {
    __shared__ float As[FC_BM][FC_PAD];
    __shared__ float Bs[FC_BN][FC_PAD];
    const int m0 = blockIdx.y * FC_BM;
    const int n0 = blockIdx.x * FC_BN;
    const int tid = threadIdx.x;
    const int wv = tid >> 5, lane = tid & 31;
    const int lr = tid >> 1;            // 0..127: A row / W row
    const int lk = (tid & 1) * 8;       // 0 or 8
    v8f acc[8];
#pragma unroll
    for (int j = 0; j < 8; ++j) acc[j] = (v8f){};

    const int arow = wv * 16 + (lane & 15);
    const int kh = (lane >> 4) * 2;

    for (int kc = 0; kc < H_; kc += FC_BK) {
        const float4* ap = (const float4*)(A + (size_t)(m0 + lr) * H_ + kc + lk);
        const float4* bp = (const float4*)(W + (size_t)(n0 + lr) * H_ + kc + lk);
        float4 a0 = ap[0], a1 = ap[1];
        float4 b0 = bp[0], b1 = bp[1];
        *(float4*)&As[lr][lk]     = a0;
        *(float4*)&As[lr][lk + 4] = a1;
        *(float4*)&Bs[lr][lk]     = b0;
        *(float4*)&Bs[lr][lk + 4] = b1;
        __syncthreads();
#pragma unroll
        for (int kk = 0; kk < FC_BK; kk += 4) {
            v2f a = *(const v2f*)&As[arow][kk + kh];
#pragma unroll
            for (int j = 0; j < 8; ++j) {
                v2f b = *(const v2f*)&Bs[j * 16 + (lane & 15)][kk + kh];
                acc[j] = wmma_k4(a, b, acc[j]);
            }
        }
        __syncthreads();
    }
    const int mbase = m0 + wv * 16 + (lane >> 4) * 8;
#pragma unroll
    for (int j = 0; j < 8; ++j) {
        int n = n0 + j * 16 + (lane & 15);
        float bn = bias[n];
#pragma unroll
        for (int r = 0; r < 8; ++r)
            Cout[(size_t)(mbase + r) * V_ + n] = acc[j][r] + bn;
    }
}

// ---------------- launcher ----------------

extern "C" void kernel_launch(void* const* d_in, const int* in_sizes, int n_in,
                              void* d_out, int out_size, void* d_ws, size_t ws_size,
                              hipStream_t stream) {
    const int*   input_seq = (const int*)  d_in[0];
    const float* midi      = (const float*)d_in[1];
    const float* emb       = (const float*)d_in[2];
    const float* mp_w      = (const float*)d_in[3];
    const float* mp_b      = (const float*)d_in[4];
    const float* wh        = (const float*)d_in[5];
    const float* wm        = (const float*)d_in[6];
    const float* vvec      = (const float*)d_in[7];
    const float* w_ih0     = (const float*)d_in[8];
    const float* w_hh0     = (const float*)d_in[9];
    const float* b_ih0     = (const float*)d_in[10];
    const float* b_hh0     = (const float*)d_in[11];
    const float* w_ih1     = (const float*)d_in[12];
    const float* w_hh1     = (const float*)d_in[13];
    const float* b_ih1     = (const float*)d_in[14];
    const float* b_hh1     = (const float*)d_in[15];
    const float* fc_w      = (const float*)d_in[16];
    const float* fc_b      = (const float*)d_in[17];

    float* out      = (float*)d_out;
    float* logits   = out;                                   // [B,T,V]
    float* outF     = out + (size_t)B_ * T_ * V_;            // hF [2,B,H] then cF [2,B,H]
    float* out_attn = outF + 4 * B_ * H_;                    // [B,T,NF]

    float* ws = (float*)d_ws;
    float* ws_emb   = ws;                      // 614400
    float* ws_mproj = ws_emb   + 614400;       // 131072
    float* ws_mkey  = ws_mproj + 131072;       // 2097152
    float* ws_q     = ws_mkey  + 2097152;      // 16384
    float* ws_sc    = ws_q     + 16384;        // 4096
    float* ws_ctx   = ws_sc    + 4096;         // 1024
    float* ws_h0    = ws_ctx   + 1024;         // 16384
    float* ws_c0    = ws_h0    + 16384;        // 16384
    float* ws_h1    = ws_c0    + 16384;        // 16384
    float* ws_c1    = ws_h1    + 16384;        // 16384
    float* ws_g0    = ws_c1    + 16384;        // 65536
    float* ws_g1    = ws_g0    + 65536;        // 65536
    float* ws_lstm  = ws_g1    + 65536;        // 1048576

    // zero recurrent state (h0,c0,h1,c1 are contiguous)
    k_zero<<<256, 256, 0, stream>>>(ws_h0, 4 * B_ * H_);
    // gather embeddings
    k_embed<<<2400, 256, 0, stream>>>(input_seq, emb, ws_emb);
    // midi projection + key (WMMA)
    k_midi_proj<<<64, 256, 0, stream>>>(midi, mp_w, mp_b, ws_mproj);
    k_midi_key<<<1024, 256, 0, stream>>>(ws_mproj, wm, ws_mkey);

    for (int t = 0; t < T_; ++t) {
        k_q<<<8, 256, 0, stream>>>(ws_h1, wh, ws_q);
        k_scores<<<512, 256, 0, stream>>>(ws_q, ws_mkey, vvec, ws_sc);
        k_softmax_ctx<<<32, 128, 0, stream>>>(ws_sc, ws_mproj, ws_ctx, out_attn, t);
        k_gates0<<<32, 256, 0, stream>>>(ws_emb, t, ws_ctx, ws_h0,
                                         w_ih0, w_hh0, b_ih0, b_hh0, ws_g0);
        k_cell<<<64, 256, 0, stream>>>(ws_g0, ws_h0, ws_c0, (float*)nullptr, t);
        k_gates1<<<32, 256, 0, stream>>>(ws_h0, ws_h1, w_ih1, w_hh1, b_ih1, b_hh1, ws_g1);
        k_cell<<<64, 256, 0, stream>>>(ws_g1, ws_h1, ws_c1, ws_lstm, t);
    }

    k_final_state<<<64, 256, 0, stream>>>(ws_h0, ws_c0, ws_h1, ws_c1, outF);
    k_fc<<<dim3(V_ / FC_BN, (B_ * T_) / FC_BM), 256, 0, stream>>>(ws_lstm, fc_w, fc_b, logits);
}